// Policy_11089605558567
// MI455X (gfx1250) — compile-verified
//
#include <hip/hip_runtime.h>
#include <math.h>

// MI455X (gfx1250) policy-logp kernel.
// One 256-thread workgroup (8 wave32) per group g (2500 groups).
// Heavy einsums run on V_WMMA_F32_16X16X4_F32 with LDS-staged operands.

typedef float v2f __attribute__((ext_vector_type(2)));
typedef float v8f __attribute__((ext_vector_type(8)));

#define G_CNT 2500
#define PER   20
#define COVD  1152
#define INVD  384
#define WID   128

__device__ __forceinline__ v8f wmma4(v2f a, v2f b, v8f c) {
  // D = A(16x4) x B(4x16) + C, fp32. args: (neg_a, A, neg_b, B, c_mod, C, reuse_a, reuse_b)
  return __builtin_amdgcn_wmma_f32_16x16x4_f32(false, a, false, b, (short)0, c, false, false);
}

__device__ __forceinline__ float wred_sum(float v) {
#pragma unroll
  for (int off = 16; off > 0; off >>= 1) v += __shfl_xor(v, off, 32);
  return v;
}

__global__ __launch_bounds__(256) void policy_kernel(
    const float* __restrict__ s_inter, const float* __restrict__ bag,
    const int* __restrict__ focus, const int* __restrict__ element,
    const float* __restrict__ distance, const float* __restrict__ orientation,
    const float* __restrict__ W_bag, const float* __restrict__ d_log_stds,
    const float* __restrict__ W_mix,
    const float* __restrict__ W1f, const float* __restrict__ b1f,
    const float* __restrict__ W2f, const float* __restrict__ b2f,
    const float* __restrict__ W1e, const float* __restrict__ b1e,
    const float* __restrict__ W2e, const float* __restrict__ b2e,
    const float* __restrict__ W1d, const float* __restrict__ b1d,
    const float* __restrict__ W2d, const float* __restrict__ b2d,
    float* __restrict__ out)
{
  extern __shared__ float sm[];
  const int SA = 136, SX = 112, SC = 112, SSV = 388;
  float* A_sm = sm;              // 128*136 = 17408  ATl stored [i][o]
  float* X_sm = sm + 17408;      // 128*112 = 14336  Xl stored [i][col]
  float* C_sm = sm + 31744;      // 128*112 = 14336  Cl stored [o][col]
  float* H1f  = C_sm;            // 20*128 (aliases C after cov phase)
  float* H1e  = C_sm + 4096;     // 20*128
  float* SV   = sm + 46080;      // 20*388           s_inv rows
  float* FOC  = sm + 53840;      // 1152             focus sample s_cov
  float* SCR  = sm + 54992;      // 512 scratch
  float* fl_s  = SCR;            // 20
  float* fp_s  = SCR + 20;       // 20
  float* hel_s = SCR + 40;       // 20
  float* el_s  = SCR + 64;       // 80
  float* cond  = SCR + 144;      // 9
  float* h1d   = SCR + 160;      // 128
  float* out6  = SCR + 288;      // 6
  float* redm  = SCR + 296;      // 8
  float* reds  = SCR + 304;      // 8
  float* sc    = SCR + 312;      // misc scalars

  const int g    = blockIdx.x;
  const int tid  = threadIdx.x;
  const int lane = tid & 31, wave = tid >> 5;
  const int half = lane >> 4, l16 = lane & 15;

  const int   focg = focus[g];
  const int   eleg = element[g];
  const float dist = distance[g];
  const float bag0 = bag[g*4+0], bag1 = bag[g*4+1], bag2 = bag[g*4+2], bag3 = bag[g*4+3];
  const float INVSQ = 0.04419417382415922f;   // 1/sqrt(512)

  const int cl[3]   = {1, 3, 5};
  const int offl[3] = {0, 128, 512};
  const int ctl[3]  = {2, 4, 7};              // ceil(20*c/16)

  // ---- per-l: build A, stage X, WMMA GEMM, reduce to s_inv + focus cov ----
  for (int l = 0; l < 3; ++l) {
    const int c = cl[l], off = offl[l], cols = PER * c, nct = ctl[l];

    // P1: AT[o][i] = sum_z bag[z]*W_bag[l][i][z][o], stored A_sm[i*SA+o]
    for (int e = tid; e < 128 * 128; e += 256) {
      int o = e & 127, i = e >> 7;
      const float* wb = W_bag + ((size_t)((l * 128 + i) * 4)) * 128 + o;
      A_sm[i * SA + o] = bag0 * wb[0] + bag1 * wb[128] + bag2 * wb[256] + bag3 * wb[384];
    }
    // P2: X_sm[i][s*c+m] from s_inter
    const int tot = 128 * c * PER;
    for (int e = tid; e < tot; e += 256) {
      int s = e / (128 * c);
      int j = e - s * (128 * c);        // j = i*c + m, contiguous in global
      int i = j / c, m = j - i * c;
      X_sm[i * SX + s * c + m] = s_inter[(size_t)(g * PER + s) * COVD + off + j];
    }
    __syncthreads();

    // P3: C[o,col] = sum_i AT[o][i] * X[i][col]  (wave w owns o-tile w)
    for (int ct = 0; ct < nct; ++ct) {
      const int colb = ct * 16;
      v8f acc = {0.f, 0.f, 0.f, 0.f, 0.f, 0.f, 0.f, 0.f};
      for (int k = 0; k < 128; k += 4) {
        const int ka = k + half * 2;
        v2f a, b;
        a.x = A_sm[ka * SA + wave * 16 + l16];
        a.y = A_sm[(ka + 1) * SA + wave * 16 + l16];
        b.x = X_sm[ka * SX + colb + l16];
        b.y = X_sm[(ka + 1) * SX + colb + l16];
        acc = wmma4(a, b, acc);
      }
      const int col = colb + l16;
      if (col < cols) {
#pragma unroll
        for (int r = 0; r < 8; ++r)
          C_sm[(wave * 16 + r + half * 8) * SC + col] = acc[r] * INVSQ;
      }
    }
    __syncthreads();

    // P4: s_inv[s][l*128+o] = sqrt(sum_m C^2 + 1e-12); extract focus cov
    for (int e = tid; e < PER * 128; e += 256) {
      int s = e >> 7, o = e & 127;
      float a2 = 1e-12f;
      for (int m = 0; m < c; ++m) { float v = C_sm[o * SC + s * c + m]; a2 += v * v; }
      SV[s * SSV + l * 128 + o] = sqrtf(a2);
      if (s == focg)
        for (int m = 0; m < c; ++m) FOC[off + o * c + m] = C_sm[o * SC + s * c + m];
    }
    __syncthreads();
  }

  // ---- P5: hidden layers of f-net / e-net via WMMA (M=20 pad 32, K=384, N=128) ----
  for (int net = 0; net < 2; ++net) {
    const float* W1 = net ? W1e : W1f;
    const float* b1 = net ? b1e : b1f;
    float* H1 = net ? H1e : H1f;
    for (int mt = 0; mt < 2; ++mt) {
      int srow = mt * 16 + l16; if (srow >= PER) srow = 0;   // clamp pad rows
      const int h = wave * 16 + l16;
      v8f acc = {0.f, 0.f, 0.f, 0.f, 0.f, 0.f, 0.f, 0.f};
      for (int k = 0; k < INVD; k += 4) {
        const int ka = k + half * 2;
        v2f a, b;
        a.x = SV[srow * SSV + ka];
        a.y = SV[srow * SSV + ka + 1];
        b.x = W1[ka * WID + h];
        b.y = W1[(ka + 1) * WID + h];
        acc = wmma4(a, b, acc);
      }
#pragma unroll
      for (int r = 0; r < 8; ++r) {
        int s = mt * 16 + r + half * 8;
        if (s < PER) H1[s * WID + h] = fmaxf(acc[r] + b1[h], 0.f);
      }
    }
  }
  __syncthreads();

  // ---- P6: fl[s], el[s][z] (second layers, wave-parallel over samples) ----
  for (int s = wave; s < PER; s += 8) {
    float af = 0.f, a0 = 0.f, a1 = 0.f, a2 = 0.f, a3 = 0.f;
    for (int hh = lane; hh < WID; hh += 32) {
      float vf = H1f[s * WID + hh];
      float ve = H1e[s * WID + hh];
      af += vf * W2f[hh];
      a0 += ve * W2e[hh * 4 + 0];
      a1 += ve * W2e[hh * 4 + 1];
      a2 += ve * W2e[hh * 4 + 2];
      a3 += ve * W2e[hh * 4 + 3];
    }
    af = wred_sum(af); a0 = wred_sum(a0); a1 = wred_sum(a1);
    a2 = wred_sum(a2); a3 = wred_sum(a3);
    if (lane == 0) {
      fl_s[s] = af + b2f[0];
      el_s[s * 4 + 0] = a0 + b2e[0];
      el_s[s * 4 + 1] = a1 + b2e[1];
      el_s[s * 4 + 2] = a2 + b2e[2];
      el_s[s * 4 + 3] = a3 + b2e[3];
    }
  }

  // ---- P7: cond_cov (Bessel-mixed) + distance-head hidden layer ----
  float bess[8];
  {
    const float cb = 1.0540925533894598f;        // sqrt(2/1.8)
#pragma unroll
    for (int k = 0; k < 8; ++k)
      bess[k] = cb * sinf((float)(k + 1) * 1.7453292519943295f * dist) / dist;
  }
  for (int p = wave; p < 9; p += 8) {
    int l, m;
    if (p == 0)      { l = 0; m = 0; }
    else if (p < 4)  { l = 1; m = p - 1; }
    else             { l = 2; m = p - 4; }
    const int c = cl[l], off = offl[l];
    float acc = 0.f;
    for (int i = lane; i < 128; i += 32) {
      float tp = 0.f;
      const int cidx = (l * 128 + i) * 4 + eleg;
#pragma unroll
      for (int k = 0; k < 8; ++k) tp += bess[k] * W_mix[k * 1536 + cidx];
      acc += FOC[off + i * c + m] * tp;
    }
    acc = wred_sum(acc);
    if (lane == 0) cond[p] = acc * 0.35355339059327373f * INVSQ;  // /sqrt(8)/sqrt(512)
  }
  if (tid < WID) {
    float a = b1d[tid] + W1d[(INVD + eleg) * WID + tid];          // one-hot row
    for (int k = 0; k < INVD; ++k) a += SV[focg * SSV + k] * W1d[k * WID + tid];
    h1d[tid] = fmaxf(a, 0.f);
  }
  __syncthreads();

  // ---- P8: head outputs, per-sample element softmax, focus softmax ----
  if (tid < 6) {
    float a = b2d[tid];
    for (int hh = 0; hh < WID; ++hh) a += h1d[hh] * W2d[hh * 6 + tid];
    out6[tid] = a;
  }
  if (tid < PER) {
    float lz[4]; float mx = -3.0e38f;
#pragma unroll
    for (int z = 0; z < 4; ++z) {
      lz[z] = (bag[g * 4 + z] > 0.f) ? el_s[tid * 4 + z] : -1.0e9f;
      mx = fmaxf(mx, lz[z]);
    }
    float se = 0.f;
#pragma unroll
    for (int z = 0; z < 4; ++z) { lz[z] = expf(lz[z] - mx); se += lz[z]; }
    float hh = 0.f;
#pragma unroll
    for (int z = 0; z < 4; ++z) {
      float pp = lz[z] / se;
      hh -= pp * logf(fmaxf(pp, 1e-20f));
    }
    hel_s[tid] = hh;
    if (tid == focg) sc[2] = logf(lz[eleg] / se + 1e-20f);        // lp_element
  }
  if (tid == 0) {
    float mx = -3.0e38f;
    for (int s = 0; s < PER; ++s) mx = fmaxf(mx, fl_s[s]);
    float se = 0.f;
    for (int s = 0; s < PER; ++s) { float e = expf(fl_s[s] - mx); fp_s[s] = e; se += e; }
    float inv = 1.f / se, hfoc = 0.f;
    for (int s = 0; s < PER; ++s) {
      float p = fp_s[s] * inv; fp_s[s] = p;
      hfoc -= p * logf(p + 1e-20f);
    }
    sc[0] = logf(fp_s[focg] + 1e-20f);                            // lp_focus
    sc[1] = hfoc;
  }
  __syncthreads();

  // ---- P9: thread0 scalars (GMM distance, entropy, f_x); all threads grid LSE ----
  if (tid == 0) {
    float l0 = out6[0], l1 = out6[1], l2 = out6[2];
    float mx = fmaxf(l0, fmaxf(l1, l2));
    float lse = logf(expf(l0 - mx) + expf(l1 - mx) + expf(l2 - mx)) + mx;
    float tm = -3.0e38f, ts = 0.f;
#pragma unroll
    for (int k = 0; k < 3; ++k) {
      float mean = tanhf(out6[3 + k]) * 0.45f + 1.35f;
      float sd = fmaxf(expf(d_log_stds[k]), 1e-6f);
      float zz = (dist - mean) / sd;
      float t = (out6[k] - lse) - 0.5f * zz * zz - logf(sd) - 0.9189385332046727f;
      if (t > tm) { ts = ts * expf(tm - t) + 1.f; tm = t; }
      else ts += expf(t - tm);
    }
    sc[3] = logf(ts) + tm;                                        // lp_dist
    float hef = 0.f;
    for (int s = 0; s < PER; ++s) hef += fp_s[s] * hel_s[s];
    sc[4] = sc[1] + hef;                                          // entropy
    float ox = orientation[g * 3 + 0], oy = orientation[g * 3 + 1], oz = orientation[g * 3 + 2];
    float nn = 1.f / sqrtf(ox * ox + oy * oy + oz * oz);
    ox *= nn; oy *= nn; oz *= nn;
    float fx = cond[0] * 0.28209479177387814f
             + cond[1] * (0.4886025119029199f * oy)
             + cond[2] * (0.4886025119029199f * oz)
             + cond[3] * (0.4886025119029199f * ox)
             + cond[4] * (1.0925484305920792f * ox * oy)
             + cond[5] * (1.0925484305920792f * oy * oz)
             + cond[6] * (0.31539156525252005f * (3.f * oz * oz - 1.f))
             + cond[7] * (1.0925484305920792f * ox * oz)
             + cond[8] * (0.5462742152960396f * (ox * ox - oy * oy));
    sc[5] = fx;
  }
  float cj[9];
#pragma unroll
  for (int j = 0; j < 9; ++j) cj[j] = cond[j];
  float gm = -3.0e38f, gs = 0.f;
  for (int i = tid; i < 4096; i += 256) {
    double ph = 2.399963229728653 * (double)i;                    // pi*(3-sqrt(5))
    double zd = 1.0 - (2.0 * (double)i + 1.0) / 4096.0;
    double rd = sqrt(fmax(1.0 - zd * zd, 0.0));
    float x = (float)(rd * cos(ph));
    float y = (float)(rd * sin(ph));
    float z = (float)zd;
    float f = cj[0] * 0.28209479177387814f
            + cj[1] * (0.4886025119029199f * y)
            + cj[2] * (0.4886025119029199f * z)
            + cj[3] * (0.4886025119029199f * x)
            + cj[4] * (1.0925484305920792f * x * y)
            + cj[5] * (1.0925484305920792f * y * z)
            + cj[6] * (0.31539156525252005f * (3.f * z * z - 1.f))
            + cj[7] * (1.0925484305920792f * x * z)
            + cj[8] * (0.5462742152960396f * (x * x - y * y));
    float t = 10.0f * f;
    if (t > gm) { gs = gs * expf(gm - t) + 1.f; gm = t; }
    else gs += expf(t - gm);
  }
#pragma unroll
  for (int off2 = 16; off2 > 0; off2 >>= 1) {
    float m2 = __shfl_xor(gm, off2, 32);
    float s2 = __shfl_xor(gs, off2, 32);
    float mn = fmaxf(gm, m2);
    gs = gs * expf(gm - mn) + s2 * expf(m2 - mn);
    gm = mn;
  }
  if (lane == 0) { redm[wave] = gm; reds[wave] = gs; }
  __syncthreads();

  if (tid == 0) {
    float M = -3.0e38f;
    for (int w = 0; w < 8; ++w) M = fmaxf(M, redm[w]);
    float S = 0.f;
    for (int w = 0; w < 8; ++w) S += reds[w] * expf(redm[w] - M);
    float logZ = logf(S) + M - 8.317766166719343f + 2.5310242469692907f; // -log(4096)+log(4pi)
    out[(size_t)g * 2 + 0] = sc[0] + sc[2] + sc[3] + 10.0f * sc[5] - logZ;
    out[(size_t)g * 2 + 1] = sc[4];
  }
}

extern "C" void kernel_launch(void* const* d_in, const int* in_sizes, int n_in,
                              void* d_out, int out_size, void* d_ws, size_t ws_size,
                              hipStream_t stream) {
  (void)in_sizes; (void)n_in; (void)d_ws; (void)ws_size; (void)out_size;
  const float* s_inter     = (const float*)d_in[0];
  const float* bag         = (const float*)d_in[1];
  // d_in[2] = batch, d_in[3] = ptr : regular layout (batch = n/20, ptr = 20*g) — not needed
  const int*   focus       = (const int*)d_in[4];
  const int*   element     = (const int*)d_in[5];
  const float* distance    = (const float*)d_in[6];
  const float* orientation = (const float*)d_in[7];
  const float* W_bag       = (const float*)d_in[8];
  const float* d_log_stds  = (const float*)d_in[9];
  const float* W_mix       = (const float*)d_in[10];
  const float* W1f = (const float*)d_in[11];
  const float* b1f = (const float*)d_in[12];
  const float* W2f = (const float*)d_in[13];
  const float* b2f = (const float*)d_in[14];
  const float* W1e = (const float*)d_in[15];
  const float* b1e = (const float*)d_in[16];
  const float* W2e = (const float*)d_in[17];
  const float* b2e = (const float*)d_in[18];
  const float* W1d = (const float*)d_in[19];
  const float* b1d = (const float*)d_in[20];
  const float* W2d = (const float*)d_in[21];
  const float* b2d = (const float*)d_in[22];
  float* out = (float*)d_out;

  const size_t shmem = 55504u * sizeof(float);   // ~217 KB dynamic LDS (< 320 KB/WGP)
  policy_kernel<<<G_CNT, 256, shmem, stream>>>(
      s_inter, bag, focus, element, distance, orientation, W_bag, d_log_stds, W_mix,
      W1f, b1f, W2f, b2f, W1e, b1e, W2e, b2e, W1d, b1d, W2d, b2d, out);
}